// CompGraphConv_71021579206963
// MI455X (gfx1250) — compile-verified
//
#include <hip/hip_runtime.h>
#include <stdint.h>

// CDNA5 / gfx1250: wave32, WMMA f32 16x16x4 path.
typedef float v2f __attribute__((ext_vector_type(2)));
typedef float v8f __attribute__((ext_vector_type(8)));

#define D_DIM 64
#define N_NODES_C 100000
#define N_EDGES_C 1280000

__device__ __forceinline__ v8f wmma_f32(v2f a, v2f b, v8f c) {
  // 8 args: (neg_a, A, neg_b, B, c_mod, C, reuse_a, reuse_b)
  return __builtin_amdgcn_wmma_f32_16x16x4_f32(false, a, false, b, (short)0, c,
                                               false, false);
}

// One wave handles one 16-edge tile: computes
//   comp = node_feat[src] * edge_feat          (A fragments, on the fly)
//   msg_fwd = comp @ W_in ; msg_rev = comp @ W_out ; edge_out = edge_feat @ W_rel
// then per-edge select/norm + atomic scatter-add into agg (node region of d_out).
__global__ __launch_bounds__(256) void edge_kernel(
    const float* __restrict__ node_feat, const float* __restrict__ edge_feat,
    const float* __restrict__ edge_norm, const int* __restrict__ src,
    const int* __restrict__ dst, const uint8_t* __restrict__ is_rev,
    const float* __restrict__ w_in, const float* __restrict__ w_out,
    const float* __restrict__ w_rel, float* __restrict__ agg,
    float* __restrict__ edge_out) {
  __shared__ float sIn[D_DIM * D_DIM];
  __shared__ float sOut[D_DIM * D_DIM];
  __shared__ float sRel[D_DIM * D_DIM];
  // Cooperative weight stage: 3 x 16KB into LDS (WGP has 320KB).
  for (int i = threadIdx.x; i < D_DIM * D_DIM / 4; i += blockDim.x) {
    ((float4*)sIn)[i] = ((const float4*)w_in)[i];
    ((float4*)sOut)[i] = ((const float4*)w_out)[i];
    ((float4*)sRel)[i] = ((const float4*)w_rel)[i];
  }
  __syncthreads();

  const int wave = threadIdx.x >> 5;
  const int lane = threadIdx.x & 31;
  const int n = lane & 15;   // A-row / C-col lane index
  const int h = lane >> 4;   // half-wave select
  const int tile = blockIdx.x * 8 + wave;  // 1280000/16 = 80000 tiles exactly

  const int eA = tile * 16 + n;  // edge supplying this lane's A row
  const int srow = src[eA];
  const float* __restrict__ efRow = edge_feat + (size_t)eA * D_DIM;
  const float* __restrict__ nfRow = node_feat + (size_t)srow * D_DIM;

  for (int t = 0; t < 4; ++t) {  // four 16-wide N tiles of D_H=64
    v8f accF = {};
    v8f accR = {};
    v8f accE = {};
    const int col = t * 16 + n;
#pragma unroll
    for (int s = 0; s < 16; ++s) {  // K = 64 in steps of 4
      const int k = 4 * s + 2 * h;  // A and B lane-local K base coincide
      v2f ae, an, bIn, bOut, bRel;
      ae.x = efRow[k];
      ae.y = efRow[k + 1];
      an.x = nfRow[k];
      an.y = nfRow[k + 1];
      v2f comp = ae * an;
      bIn.x = sIn[k * D_DIM + col];
      bIn.y = sIn[(k + 1) * D_DIM + col];
      bOut.x = sOut[k * D_DIM + col];
      bOut.y = sOut[(k + 1) * D_DIM + col];
      bRel.x = sRel[k * D_DIM + col];
      bRel.y = sRel[(k + 1) * D_DIM + col];
      accF = wmma_f32(comp, bIn, accF);
      accR = wmma_f32(comp, bOut, accR);
      accE = wmma_f32(ae, bRel, accE);
    }
    // Epilogue: C/D layout -> lane(n,h), vgpr r holds [M = r + 8h][N = n].
#pragma unroll
    for (int r = 0; r < 8; ++r) {
      const int e = tile * 16 + r + 8 * h;
      const float nrm = edge_norm[e];
      const float m = (is_rev[e] ? accR[r] : accF[r]) * nrm;
      atomicAdd(&agg[(size_t)dst[e] * D_DIM + col], m);
      edge_out[(size_t)e * D_DIM + col] = accE[r];
    }
  }
}

// One wave per 16-node tile: loop_msg = (node_feat * loop_rel) @ W_loop,
// then node_out = (agg + loop_msg) / 3 + bias, in place over the agg buffer.
__global__ __launch_bounds__(256) void node_kernel(
    const float* __restrict__ node_feat, const float* __restrict__ w_loop,
    const float* __restrict__ loop_rel, const float* __restrict__ bias,
    float* __restrict__ node_out) {
  __shared__ float sW[D_DIM * D_DIM];
  __shared__ float sLR[D_DIM];
  for (int i = threadIdx.x; i < D_DIM * D_DIM / 4; i += blockDim.x)
    ((float4*)sW)[i] = ((const float4*)w_loop)[i];
  if (threadIdx.x < D_DIM / 4)
    ((float4*)sLR)[threadIdx.x] = ((const float4*)loop_rel)[threadIdx.x];
  __syncthreads();

  const int wave = threadIdx.x >> 5;
  const int lane = threadIdx.x & 31;
  const int n = lane & 15;
  const int h = lane >> 4;
  const int tile = blockIdx.x * 8 + wave;
  if (tile >= N_NODES_C / 16) return;  // wave-uniform exit, EXEC stays full

  const int row = tile * 16 + n;
  const float* __restrict__ nfRow = node_feat + (size_t)row * D_DIM;

  for (int t = 0; t < 4; ++t) {
    v8f acc = {};
    const int col = t * 16 + n;
#pragma unroll
    for (int s = 0; s < 16; ++s) {
      const int k = 4 * s + 2 * h;
      v2f a, b;
      a.x = nfRow[k] * sLR[k];
      a.y = nfRow[k + 1] * sLR[k + 1];
      b.x = sW[k * D_DIM + col];
      b.y = sW[(k + 1) * D_DIM + col];
      acc = wmma_f32(a, b, acc);
    }
    const float bv = bias[col];
#pragma unroll
    for (int r = 0; r < 8; ++r) {
      const int node = tile * 16 + r + 8 * h;
      const size_t idx = (size_t)node * D_DIM + col;
      node_out[idx] = (node_out[idx] + acc[r]) * 0.3333333f + bv;
    }
  }
}

extern "C" void kernel_launch(void* const* d_in, const int* in_sizes, int n_in,
                              void* d_out, int out_size, void* d_ws,
                              size_t ws_size, hipStream_t stream) {
  const float* node_feat = (const float*)d_in[0];
  const float* edge_feat = (const float*)d_in[1];
  const float* edge_norm = (const float*)d_in[2];
  const int* src = (const int*)d_in[3];
  const int* dst = (const int*)d_in[4];
  const uint8_t* is_rev = (const uint8_t*)d_in[5];  // jax bool_ = 1 byte
  const float* w_in = (const float*)d_in[6];
  const float* w_out = (const float*)d_in[7];
  const float* w_rel = (const float*)d_in[8];
  const float* w_loop = (const float*)d_in[9];
  const float* loop_rel = (const float*)d_in[10];
  const float* bias = (const float*)d_in[11];

  float* node_out = (float*)d_out;  // first N_NODES*64 floats
  float* edge_out = (float*)d_out + (size_t)N_NODES_C * D_DIM;

  // Zero the aggregation region every call (harness poisons d_out once).
  hipMemsetAsync(node_out, 0, (size_t)N_NODES_C * D_DIM * sizeof(float),
                 stream);

  // 80000 edge tiles / 8 waves per block = 10000 blocks exactly.
  edge_kernel<<<N_EDGES_C / 16 / 8, 256, 0, stream>>>(
      node_feat, edge_feat, edge_norm, src, dst, is_rev, w_in, w_out, w_rel,
      node_out, edge_out);

  // 6250 node tiles -> 782 blocks of 8 waves (last block partially active).
  node_kernel<<<(N_NODES_C / 16 + 7) / 8, 256, 0, stream>>>(
      node_feat, w_loop, loop_rel, bias, node_out);
}